// AreaWeightedDownsample_66305705116307
// MI455X (gfx1250) — compile-verified
//
#include <hip/hip_runtime.h>
#include <stdint.h>

// ---------------------------------------------------------------------------
// AreaWeightedDownsample for MI455X (gfx1250)
//
// out[b,p,c] = sum_{i: parent[i]==p} x[b,i,c] * omega[i] / max(sum omega, eps)
//
// HBM-bound scatter: ~0.42 GB total traffic, ~18us floor at 23.3 TB/s.
//   K3 (first in file for disasm visibility): scatter x*omega/denom with
//      async global->LDS double-buffered streaming (NT hint keeps the 84MB
//      atomic target resident in the 192MB L2)
//   K1: zero out + denom
//   K2: denom[p] += omega[i]            (f32 atomics, 160KB L2-resident)
// Normalization folded into the scatter weight -> no third pass over out.
// ---------------------------------------------------------------------------

#define WAVESZ 32
#define WPB    8            // waves per block
#define BLK    (WAVESZ*WPB) // 256 threads
#define CCH    128          // channels
#define EPSV   1e-8f

// One wave32 per (b,i) row: lane l owns float4 at channel 4*l.
// Double-buffered async global->LDS staging (CDNA5 ASYNCcnt path).
__global__ __launch_bounds__(BLK) void awd_scatter_kernel(
    const float* __restrict__ x, const float* __restrict__ omega,
    const int* __restrict__ parent, const float* __restrict__ denom,
    float* __restrict__ out, int n_in, int n_out, int total_rows) {
    __shared__ __align__(16) float smem[2][WPB][CCH];

    const int wave = threadIdx.x >> 5;
    const int lane = threadIdx.x & 31;
    const int c0   = lane << 2;                 // channel base for this lane

    const int gw      = blockIdx.x * WPB + wave;
    const int wstride = gridDim.x * WPB;

    // LDS byte addresses for this lane's 16B slot in each buffer.
    // Generic->LDS: hardware uses addr[31:0] as the LDS offset (ISA 10.2).
    uint32_t lds0 = (uint32_t)(uintptr_t)&smem[0][wave][c0];
    uint32_t lds1 = (uint32_t)(uintptr_t)&smem[1][wave][c0];

    // Prologue: kick off row 0 for this wave.
    if (gw < total_rows) {
        const float* g = x + (long)gw * CCH + c0;
        asm volatile("global_load_async_to_lds_b128 %0, %1, off th:TH_LOAD_NT"
                     :: "v"(lds0), "v"(g) : "memory");
    }

    int buf = 0;
    for (int r = gw; r < total_rows; r += wstride, buf ^= 1) {
        const int rn = r + wstride;
        if (rn < total_rows) {
            // Prefetch next row into the other buffer, then wait so that
            // only the just-issued load may still be in flight (ASYNCcnt<=1
            // => row r has landed, since async loads complete in order).
            const float* g = x + (long)rn * CCH + c0;
            if (buf == 0) {
                asm volatile("global_load_async_to_lds_b128 %0, %1, off th:TH_LOAD_NT"
                             :: "v"(lds1), "v"(g) : "memory");
            } else {
                asm volatile("global_load_async_to_lds_b128 %0, %1, off th:TH_LOAD_NT"
                             :: "v"(lds0), "v"(g) : "memory");
            }
            asm volatile("s_wait_asynccnt 0x1" ::: "memory");
        } else {
            asm volatile("s_wait_asynccnt 0x0" ::: "memory");
        }

        const int b = r / n_in;            // row -> (batch, node)
        const int i = r - b * n_in;
        const int p = parent[i];
        float d = denom[p];
        d = fmaxf(d, EPSV);
        const float w = omega[i] / d;      // normalization folded in

        float4 v = *(const float4*)&smem[buf][wave][c0];

        float* o = out + ((long)b * n_out + p) * CCH + c0;
        __hip_atomic_fetch_add(o + 0, v.x * w, __ATOMIC_RELAXED, __HIP_MEMORY_SCOPE_AGENT);
        __hip_atomic_fetch_add(o + 1, v.y * w, __ATOMIC_RELAXED, __HIP_MEMORY_SCOPE_AGENT);
        __hip_atomic_fetch_add(o + 2, v.z * w, __ATOMIC_RELAXED, __HIP_MEMORY_SCOPE_AGENT);
        __hip_atomic_fetch_add(o + 3, v.w * w, __ATOMIC_RELAXED, __HIP_MEMORY_SCOPE_AGENT);
    }
}

__global__ void awd_zero_kernel(float* __restrict__ out, long n_out4,
                                float* __restrict__ denom, int n_den) {
    long i      = (long)blockIdx.x * blockDim.x + threadIdx.x;
    long stride = (long)gridDim.x * blockDim.x;
    float4 z = {0.f, 0.f, 0.f, 0.f};
    for (long k = i; k < n_out4; k += stride) ((float4*)out)[k] = z;
    for (long k = i; k < n_den;  k += stride) denom[k] = 0.f;
}

__global__ void awd_denom_kernel(const float* __restrict__ omega,
                                 const int*   __restrict__ parent,
                                 float*       __restrict__ denom, int n_in) {
    int i = blockIdx.x * blockDim.x + threadIdx.x;
    if (i < n_in) {
        __hip_atomic_fetch_add(&denom[parent[i]], omega[i],
                               __ATOMIC_RELAXED, __HIP_MEMORY_SCOPE_AGENT);
    }
}

extern "C" void kernel_launch(void* const* d_in, const int* in_sizes, int n_in_args,
                              void* d_out, int out_size, void* d_ws, size_t ws_size,
                              hipStream_t stream) {
    const float* x     = (const float*)d_in[0];  // [B, N_in, C] f32
    const float* omega = (const float*)d_in[1];  // [N_in] f32
    const int*  parent = (const int*)d_in[2];    // [N_in] int
    (void)n_in_args; (void)ws_size;

    const int C    = 128;
    const int n_in = in_sizes[1];
    const int B    = in_sizes[0] / (n_in * C);
    const int n_out = out_size / (B * C);

    float* out   = (float*)d_out;
    float* denom = (float*)d_ws;                 // [n_out] f32 scratch

    // K1: zero accumulation targets (harness poisons d_out/d_ws with 0xAA).
    awd_zero_kernel<<<2048, 256, 0, stream>>>(out, (long)out_size / 4, denom, n_out);

    // K2: denom[p] = sum omega over children.
    awd_denom_kernel<<<(n_in + 255) / 256, 256, 0, stream>>>(omega, parent, denom, n_in);

    // K3: scatter-accumulate normalized weighted features.
    const int total_rows = B * n_in;             // 655368
    awd_scatter_kernel<<<4096, BLK, 0, stream>>>(x, omega, parent, denom, out,
                                                 n_in, n_out, total_rows);
}